// VMSplitFeatureVolume_73761768342118
// MI455X (gfx1250) — compile-verified
//
#include <hip/hip_runtime.h>
#include <hip/hip_bf16.h>

#define RES      300
#define NDC      16          // density components
#define NAC      48          // appearance components
#define APPD     27          // output appearance dim
#define KDIM     (3*NAC)     // 144
#define PLANE    (RES*RES)
#define WPB      2           // waves per block
#define BLOCK    (WPB*32)
#define BRES     32          // basis LDS rows (27 real + 5 zero-padded -> no masking in hot loop)
#define BROW     145         // basis LDS row stride (145 % 64 = 17, odd -> conflict-free column reads)
#define FROW     40          // feats LDS row stride (half-wave row offset -> +16 banks, disjoint ranges)

typedef __attribute__((ext_vector_type(2))) float v2f;
typedef __attribute__((ext_vector_type(8))) float v8f;

__global__ __launch_bounds__(BLOCK)
void vmsplit_fused_kernel(const float* __restrict__ xyz,
                          const float* __restrict__ dpl,   // (3,16,300,300)
                          const float* __restrict__ dln,   // (3,16,300)
                          const float* __restrict__ apl,   // (3,48,300,300)
                          const float* __restrict__ aln,   // (3,48,300)
                          const float* __restrict__ basis, // (27,144)
                          float* __restrict__ out,         // [n] sigma ++ [n,27] app
                          int n)
{
    __shared__ float basisLds[BRES * BROW];          // 18,560 B (rows >= 27 zeroed)
    __shared__ float featsLds[WPB * KDIM * FROW];    // 46,080 B

    const int tid = threadIdx.x;

    // Stage basis_w into bank-padded LDS; zero-fill pad rows 27..31 so the
    // M-tile-1 A-fragment loads need no EXEC masking in the WMMA loop.
    for (int i = tid; i < BRES * KDIM; i += BLOCK) {
        int d = i / KDIM;
        int c = i - d * KDIM;
        basisLds[d * BROW + c] = (d < APPD) ? basis[i] : 0.0f;
    }
    __syncthreads();

    const int wave = tid >> 5;
    const int lane = tid & 31;
    const int w = blockIdx.x * WPB + wave;
    const int tileBase = w * 32;
    if (tileBase >= n) return;                       // wave-uniform

    float* fw = &featsLds[wave * KDIM * FROW];

    // ---------------- Phase A: interpolation (1 lane = 1 point) ----------------
    const int p = min(tileBase + lane, n - 1);
    const float X = xyz[3 * p + 0];
    const float Y = xyz[3 * p + 1];
    const float Z = xyz[3 * p + 2];
    const float cxa[3] = {X, X, Y};
    const float cya[3] = {Y, Z, Z};
    const float cza[3] = {Z, Y, X};

    const float S = 0.5f * (float)(RES - 1);
    float sigma = 0.0f;

    for (int o = 0; o < 3; ++o) {
        float gx = fminf(fmaxf((cxa[o] + 1.0f) * S, 0.0f), (float)(RES - 1));
        float gy = fminf(fmaxf((cya[o] + 1.0f) * S, 0.0f), (float)(RES - 1));
        float gz = fminf(fmaxf((cza[o] + 1.0f) * S, 0.0f), (float)(RES - 1));
        int ix = min((int)gx, RES - 2);              // gx >= 0 -> trunc == floor
        int iy = min((int)gy, RES - 2);
        int iz = min((int)gz, RES - 2);
        const float wx = gx - (float)ix;
        const float wy = gy - (float)iy;
        const float wz = gz - (float)iz;
        const int o00 = iy * RES + ix;
        const int o10 = o00 + RES;

        // density: sigma accumulation
        {
            const float* P = dpl + (size_t)o * NDC * PLANE;
            const float* L = dln + (size_t)o * NDC * RES;
#pragma unroll 4
            for (int c = 0; c < NDC; ++c) {
                const float* pc = P + (size_t)c * PLANE;
                float v00 = pc[o00], v01 = pc[o00 + 1];
                float v10 = pc[o10], v11 = pc[o10 + 1];
                float top = v00 + wx * (v01 - v00);
                float bot = v10 + wx * (v11 - v10);
                float pv  = top + wy * (bot - top);
                const float* lc = L + (size_t)c * RES + iz;
                float lv = lc[0] + wz * (lc[1] - lc[0]);
                sigma += pv * lv;
            }
        }
        // appearance: feats -> LDS [c][n]
        {
            const float* P = apl + (size_t)o * NAC * PLANE;
            const float* L = aln + (size_t)o * NAC * RES;
#pragma unroll 4
            for (int c = 0; c < NAC; ++c) {
                const float* pc = P + (size_t)c * PLANE;
                float v00 = pc[o00], v01 = pc[o00 + 1];
                float v10 = pc[o10], v11 = pc[o10 + 1];
                float top = v00 + wx * (v01 - v00);
                float bot = v10 + wx * (v11 - v10);
                float pv  = top + wy * (bot - top);
                const float* lc = L + (size_t)c * RES + iz;
                float lv = lc[0] + wz * (lc[1] - lc[0]);
                fw[(o * NAC + c) * FROW + lane] = pv * lv;
            }
        }
    }
    if (tileBase + lane < n) out[tileBase + lane] = sigma;

    // Same-wave LDS producer->consumer: DS ops are in-order per wave; just stop
    // the compiler from reordering across this point.
    asm volatile("" ::: "memory");

    // ---------------- Phase B: 27x144 @ 144x32 GEMM via WMMA f32 16x16x4 -------
    const int h = lane >> 4;     // half-wave selects K pair {2h, 2h+1}
    const int l = lane & 15;     // M (A) / N (B,D) index within tile
    const v8f vzero = {0.f, 0.f, 0.f, 0.f, 0.f, 0.f, 0.f, 0.f};
    v8f acc00 = vzero, acc01 = vzero, acc10 = vzero, acc11 = vzero;

    const float* b0row = &basisLds[l * BROW];          // M-tile 0: rows 0..15
    const float* b1row = &basisLds[(16 + l) * BROW];   // M-tile 1: rows 16..31 (27..31 are zero)

    for (int k = 0; k < KDIM; k += 4) {
        const int ka = k + 2 * h;
        v2f a0, a1, b0, b1;
        a0.x = b0row[ka];
        a0.y = b0row[ka + 1];
        a1.x = b1row[ka];
        a1.y = b1row[ka + 1];
        b0.x = fw[ka * FROW + l];
        b0.y = fw[(ka + 1) * FROW + l];
        b1.x = fw[ka * FROW + 16 + l];
        b1.y = fw[(ka + 1) * FROW + 16 + l];
        acc00 = __builtin_amdgcn_wmma_f32_16x16x4_f32(false, a0, false, b0, (short)0, acc00, false, false);
        acc01 = __builtin_amdgcn_wmma_f32_16x16x4_f32(false, a0, false, b1, (short)0, acc01, false, false);
        acc10 = __builtin_amdgcn_wmma_f32_16x16x4_f32(false, a1, false, b0, (short)0, acc10, false, false);
        acc11 = __builtin_amdgcn_wmma_f32_16x16x4_f32(false, a1, false, b1, (short)0, acc11, false, false);
    }

    // D layout: VGPR r <-> M = r + 8*h ; N = l. Store d<27, n<npoints.
    float* outApp = out + n;
    const int n0 = tileBase + l;
    const int n1 = tileBase + 16 + l;
#pragma unroll
    for (int r = 0; r < 8; ++r) {
        const int d0 = r + 8 * h;        // M-tile 0: d in [0,16)
        const int d1 = 16 + r + 8 * h;   // M-tile 1: d in [16,32) -> mask to <27
        if (n0 < n) {
            outApp[(size_t)n0 * APPD + d0] = acc00[r];
            if (d1 < APPD) outApp[(size_t)n0 * APPD + d1] = acc10[r];
        }
        if (n1 < n) {
            outApp[(size_t)n1 * APPD + d0] = acc01[r];
            if (d1 < APPD) outApp[(size_t)n1 * APPD + d1] = acc11[r];
        }
    }
}

extern "C" void kernel_launch(void* const* d_in, const int* in_sizes, int n_in,
                              void* d_out, int out_size, void* d_ws, size_t ws_size,
                              hipStream_t stream) {
    const float* xyz   = (const float*)d_in[0];
    const float* dpl   = (const float*)d_in[1];
    const float* dln   = (const float*)d_in[2];
    const float* apl   = (const float*)d_in[3];
    const float* aln   = (const float*)d_in[4];
    const float* basis = (const float*)d_in[5];
    float* out = (float*)d_out;

    const int n = in_sizes[0] / 3;                   // 500,000
    const int waves  = (n + 31) / 32;                // 15,625
    const int blocks = (waves + WPB - 1) / WPB;
    vmsplit_fused_kernel<<<blocks, BLOCK, 0, stream>>>(xyz, dpl, dln, apl, aln, basis, out, n);
}